// EncoderMemNN_90091234001534
// MI455X (gfx1250) — compile-verified
//
#include <hip/hip_runtime.h>

#define B_    32
#define M_    128
#define L_    32
#define H_    512
#define V_    32000
#define HOPS_ 3
#define NEG_INF_ (-1e9f)

typedef __attribute__((ext_vector_type(2))) float v2f;
typedef __attribute__((ext_vector_type(8))) float v8f;

// ---------------------------------------------------------------------------
// Kernel 1: copy enc_hidden -> running u buffer (ws)
// ---------------------------------------------------------------------------
__global__ __launch_bounds__(256) void init_u_kernel(const float* __restrict__ enc,
                                                     float* __restrict__ u_buf) {
    int i = blockIdx.x * blockDim.x + threadIdx.x;
    if (i < B_ * H_) u_buf[i] = enc[i];
}

// ---------------------------------------------------------------------------
// Kernel 2: embedding gather + pool.
// P[t][b][m][h] = sum_l C[t][ids[b][m][l]][h]      for t = 0..HOPS
// One block per (t,b,m); 128 threads x float4 = 512 = H. Coalesced b128 loads
// of each 2KB embedding row; prefetch next gathered row (global_prefetch_b8).
// ---------------------------------------------------------------------------
__global__ __launch_bounds__(128) void pool_kernel(const int* __restrict__ ids,
                                                   const float* __restrict__ C,
                                                   float* __restrict__ P) {
    int bid = blockIdx.x;            // t * (B*M) + (b*M + m); B*M = 4096
    int t   = bid >> 12;
    int rem = bid & 4095;
    int tid = threadIdx.x;

    const int*    row_ids = ids + rem * L_;
    const float4* Ct      = (const float4*)(C + (size_t)t * V_ * H_);

    float4 acc = make_float4(0.f, 0.f, 0.f, 0.f);
#pragma unroll 4
    for (int l = 0; l < L_; ++l) {
        int id = row_ids[l];
        if (l + 1 < L_) {
            int idn = row_ids[l + 1];
            __builtin_prefetch(Ct + (size_t)idn * (H_ / 4) + tid, 0, 0);
        }
        float4 v = Ct[(size_t)id * (H_ / 4) + tid];
        acc.x += v.x; acc.y += v.y; acc.z += v.z; acc.w += v.w;
    }
    ((float4*)P)[(size_t)bid * (H_ / 4) + tid] = acc;
}

// ---------------------------------------------------------------------------
// Kernel 3: one attention hop. One wave32 block per batch b.
//   scores[m] = u . P_h[b][m][:]        (WMMA f32 16x16x4, u broadcast in N)
//   attn      = softmax(mask(scores))   (wave32 shfl reductions)
//   u        += sum_m attn[m] * P_{h+1}[b][m][:]   (WMMA, attn broadcast in M)
// ---------------------------------------------------------------------------
__global__ __launch_bounds__(32) void hop_kernel(const float* __restrict__ P,
                                                 float* __restrict__ u_buf,
                                                 const int* __restrict__ lengths,
                                                 float* __restrict__ attns_out,
                                                 float* __restrict__ u_out,
                                                 int hop, int is_last) {
    __shared__ float u_sh[H_];
    __shared__ float sc_sh[M_];
    __shared__ float at_sh[M_];

    const int lane = threadIdx.x;      // 0..31
    const int grp  = lane >> 4;        // 0 or 1 (half-wave)
    const int lm   = lane & 15;
    const int b    = blockIdx.x;

    // load u[b] into LDS (4 x b128 per lane)
    {
        const float4* src = (const float4*)(u_buf + b * H_);
        float4*       dst = (float4*)u_sh;
#pragma unroll
        for (int i = 0; i < H_ / 4 / 32; ++i)
            dst[lane + i * 32] = src[lane + i * 32];
    }
    __syncthreads();

    const float* Ph = P + ((size_t)hop * B_ + b) * (M_ * H_);
    const float* Pc = P + ((size_t)(hop + 1) * B_ + b) * (M_ * H_);

    // ---- Phase 1: scores[m] = sum_k u[k] * Ph[m][k] --------------------
    // A[m][k] = Ph tile, B[k][n] = u[k] (same for every column n).
    for (int mt = 0; mt < M_ / 16; ++mt) {
        v8f acc = {};
        const float* arow = Ph + (mt * 16 + lm) * H_ + grp * 2;  // per-lane row
#pragma unroll 8
        for (int kt = 0; kt < H_ / 4; ++kt) {
            v2f a  = *(const v2f*)(arow + kt * 4);               // K = kt*4+grp*2 (+0,+1)
            v2f bb = *(const v2f*)(&u_sh[kt * 4 + grp * 2]);     // broadcast u over N
            acc = __builtin_amdgcn_wmma_f32_16x16x4_f32(
                false, a, false, bb, (short)0, acc, false, false);
        }
        // column N==0 lives in lanes with (lane&15)==0:
        //   lane 0  -> M = 0..7  (VGPR r), lane 16 -> M = 8..15
        if (lm == 0) {
#pragma unroll
            for (int r = 0; r < 8; ++r)
                sc_sh[mt * 16 + grp * 8 + r] = acc[r];
        }
    }
    __syncthreads();

    // ---- softmax over M=128 with mask (lengths==0 -> -1e9) -------------
    float s[4];
#pragma unroll
    for (int j = 0; j < 4; ++j) {
        int m = lane + j * 32;
        float v = sc_sh[m];
        if (lengths[b * M_ + m] == 0) v = NEG_INF_;
        s[j] = v;
    }
    float mx = fmaxf(fmaxf(s[0], s[1]), fmaxf(s[2], s[3]));
#pragma unroll
    for (int off = 16; off > 0; off >>= 1)
        mx = fmaxf(mx, __shfl_xor(mx, off, 32));
    float ex[4];
    float sum = 0.f;
#pragma unroll
    for (int j = 0; j < 4; ++j) { ex[j] = expf(s[j] - mx); sum += ex[j]; }
#pragma unroll
    for (int off = 16; off > 0; off >>= 1)
        sum += __shfl_xor(sum, off, 32);
    float inv = 1.f / sum;
#pragma unroll
    for (int j = 0; j < 4; ++j) {
        int m = lane + j * 32;
        float a = ex[j] * inv;
        at_sh[m] = a;
        attns_out[((size_t)hop * B_ + b) * M_ + m] = a;   // [HOPS,B,1,M]
    }
    __syncthreads();

    // ---- Phase 2: u[n] += sum_m attn[m] * Pc[m][n] ---------------------
    // A[i][k] = attn[k] (same for every row i), B[k][n] = Pc[k][n].
    for (int nt = 0; nt < H_ / 16; ++nt) {
        v8f acc = {};
#pragma unroll 4
        for (int kt = 0; kt < M_ / 4; ++kt) {
            int k0 = kt * 4 + grp * 2;
            v2f a  = *(const v2f*)(&at_sh[k0]);                  // broadcast attn over M
            v2f bb;
            bb.x = Pc[(size_t)(k0 + 0) * H_ + nt * 16 + lm];
            bb.y = Pc[(size_t)(k0 + 1) * H_ + nt * 16 + lm];
            acc = __builtin_amdgcn_wmma_f32_16x16x4_f32(
                false, a, false, bb, (short)0, acc, false, false);
        }
        // row M==0 lives in lanes 0..15, VGPR 0, N = lane
        if (grp == 0) {
            int hd = nt * 16 + lm;
            float nu = u_sh[hd] + acc[0];
            u_buf[b * H_ + hd] = nu;
            if (is_last) u_out[b * H_ + hd] = nu;
        }
    }
}

// ---------------------------------------------------------------------------
extern "C" void kernel_launch(void* const* d_in, const int* in_sizes, int n_in,
                              void* d_out, int out_size, void* d_ws, size_t ws_size,
                              hipStream_t stream) {
    const int*   inputs  = (const int*)d_in[0];    // [B,M,L]
    const int*   lengths = (const int*)d_in[1];    // [B,M]
    const float* enc     = (const float*)d_in[2];  // [B,H]
    const float* C       = (const float*)d_in[3];  // [HOPS+1,V,H]

    float* P     = (float*)d_ws;                                   // [4][B][M][H] = 32 MiB
    float* u_buf = P + (size_t)(HOPS_ + 1) * B_ * M_ * H_;         // [B][H]

    float* u_out = (float*)d_out;                 // first B*H floats
    float* attns = u_out + B_ * H_;               // then HOPS*B*1*M floats

    init_u_kernel<<<(B_ * H_ + 255) / 256, 256, 0, stream>>>(enc, u_buf);
    pool_kernel<<<(HOPS_ + 1) * B_ * M_, 128, 0, stream>>>(inputs, C, P);
    for (int h = 0; h < HOPS_; ++h)
        hop_kernel<<<B_, 32, 0, stream>>>(P, u_buf, lengths, attns, u_out,
                                          h, h == HOPS_ - 1 ? 1 : 0);
}